// Net_27891517620463
// MI455X (gfx1250) — compile-verified
//
#include <hip/hip_runtime.h>
#include <hip/hip_bf16.h>

typedef __attribute__((ext_vector_type(2))) float v2f;
typedef __attribute__((ext_vector_type(8))) float v8f;

#define N_NODES   250
#define IN_F      18
#define OUT_F     48
#define NEG_SLOPE 0.2f
#define EPS_GAT   1e-16f
#define N_EDGES   4000
#define N_TOT     4250   // edges + self loops

// ---------------------------------------------------------------------------
// Stage 1: h[250,48] = x[250,18] @ w_gat[48,18]^T  — exact fp32 WMMA.
// 16x3 tiles of 16x16, K padded to 20 (5 x v_wmma_f32_16x16x4_f32).
// One wave (32 threads) per tile; EXEC all-ones as WMMA requires.
// A 16x4 f32 layout: lanes 0-15 hold K=k0,k0+1 (v[0],v[1]); lanes 16-31 K=k0+2,k0+3.
// B 4x16 f32 layout mirrors A with N = lane%16. D: VGPR i -> (M=i(+8), N=lane%16).
// OOB handling: clamped always-in-bounds flat index + 0/1 mask multiply, so the
// compiler emits unconditional global_load_b32 (no EXEC save/branch chains).
// ---------------------------------------------------------------------------
__global__ void gat_gemm_wmma_kernel(const float* __restrict__ x,
                                     const float* __restrict__ w_gat,
                                     float* __restrict__ h) {
    const int mt   = blockIdx.x / 3;
    const int nt   = blockIdx.x % 3;
    const int m0   = mt * 16;
    const int n0   = nt * 16;
    const int lane = threadIdx.x;
    const int half = lane >> 4;
    const int l16  = lane & 15;
    const int m    = m0 + l16;   // A row for this lane
    const int o    = n0 + l16;   // B column (output feature) for this lane

    const int   aBase = min(m, N_NODES - 1) * IN_F;   // clamped row base
    const float amask = (m < N_NODES) ? 1.f : 0.f;    // row-valid mask
    const int   bBase = o * IN_F;                     // o < 48 always

    v8f acc = {0.f, 0.f, 0.f, 0.f, 0.f, 0.f, 0.f, 0.f};

#pragma unroll
    for (int k0 = 0; k0 < 20; k0 += 4) {
        const int   ka  = k0 + half * 2;
        const float km0 = (ka     < IN_F) ? 1.f : 0.f;   // K-valid masks
        const float km1 = (ka + 1 < IN_F) ? 1.f : 0.f;   // (live only at k0=16)
        v2f A, B;
        A.x = x[min(aBase + ka,     N_NODES * IN_F - 1)] * (amask * km0);
        A.y = x[min(aBase + ka + 1, N_NODES * IN_F - 1)] * (amask * km1);
        B.x = w_gat[min(bBase + ka,     OUT_F * IN_F - 1)] * km0;
        B.y = w_gat[min(bBase + ka + 1, OUT_F * IN_F - 1)] * km1;
        acc = __builtin_amdgcn_wmma_f32_16x16x4_f32(
            /*neg_a=*/false, A, /*neg_b=*/false, B,
            /*c_mod=*/(short)0, acc, /*reuse_a=*/false, /*reuse_b=*/false);
    }

#pragma unroll
    for (int i = 0; i < 8; ++i) {
        const int M = m0 + i + half * 8;
        if (M < N_NODES) h[M * OUT_F + n0 + l16] = acc[i];
    }
}

// ---------------------------------------------------------------------------
// Stage 2: a_s[n] = h[n,:]·att_src ; a_d[n] = h[n,:]·att_dst
// ---------------------------------------------------------------------------
__global__ void att_score_kernel(const float* __restrict__ h,
                                 const float* __restrict__ att_src,
                                 const float* __restrict__ att_dst,
                                 float* __restrict__ a_s,
                                 float* __restrict__ a_d) {
    const int n = blockIdx.x * blockDim.x + threadIdx.x;
    if (n >= N_NODES) return;
    float s = 0.f, d = 0.f;
#pragma unroll
    for (int o = 0; o < OUT_F; ++o) {
        const float hv = h[n * OUT_F + o];
        s += hv * att_src[o];
        d += hv * att_dst[o];
    }
    a_s[n] = s;
    a_d[n] = d;
}

// ---------------------------------------------------------------------------
// Stage 3: deterministic per-node segment softmax + aggregation + bias + relu.
// One wave per destination node; edges (incl. self loops) cached in LDS.
// out[n,o] = relu( (sum_e exp(e)-weighted h[src_e,o]) / (sum_e exp(e)+eps) + b[o] )
// ---------------------------------------------------------------------------
__global__ void gat_aggregate_kernel(const int* __restrict__ ei,
                                     const float* __restrict__ h,
                                     const float* __restrict__ a_s,
                                     const float* __restrict__ a_d,
                                     const float* __restrict__ b_gat,
                                     float* __restrict__ out) {
    __shared__ int esrc[N_TOT];
    __shared__ int edst[N_TOT];
    const int tid = threadIdx.x;
    for (int i = tid; i < N_TOT; i += blockDim.x) {
        if (i < N_EDGES) { esrc[i] = ei[i]; edst[i] = ei[N_EDGES + i]; }
        else             { esrc[i] = i - N_EDGES; edst[i] = i - N_EDGES; }
    }
    __syncthreads();

    const int wave = blockIdx.x * (blockDim.x >> 5) + (tid >> 5);
    const int lane = tid & 31;
    if (wave >= N_NODES) return;
    const int node = wave;
    const float adn = a_d[node];

    // pass 1: segment max of leaky_relu(a_s[src] + a_d[node])
    float m = -3.4e38f;
    for (int e = lane; e < N_TOT; e += 32) {
        if (edst[e] == node) {
            float v = a_s[esrc[e]] + adn;
            v = (v < 0.f) ? v * NEG_SLOPE : v;
            m = fmaxf(m, v);
        }
    }
    for (int off = 16; off; off >>= 1) m = fmaxf(m, __shfl_xor(m, off, 32));

    // pass 2: exp-weighted accumulation (lane owns feature `lane` and `lane+32`)
    float acc0 = 0.f, acc1 = 0.f, ssum = 0.f;
    const int nchunk = (N_TOT + 31) / 32;
    for (int c = 0; c < nchunk; ++c) {
        const int e = c * 32 + lane;
        int  s = 0;
        float w = 0.f;
        bool match = false;
        if (e < N_TOT && edst[e] == node) {
            match = true;
            s = esrc[e];
            float v = a_s[s] + adn;
            v = (v < 0.f) ? v * NEG_SLOPE : v;
            w = expf(v - m);
            ssum += w;
        }
        unsigned long long mask = __ballot(match);
        while (mask) {
            const int j = __ffsll(mask) - 1;
            mask &= mask - 1;
            const float wj = __shfl(w, j, 32);
            const int   sj = __shfl(s, j, 32);
            acc0 += wj * h[sj * OUT_F + lane];
            if (lane < 16) acc1 += wj * h[sj * OUT_F + 32 + lane];
        }
    }
    for (int off = 16; off; off >>= 1) ssum += __shfl_xor(ssum, off, 32);
    const float inv = 1.f / (ssum + EPS_GAT);

    out[node * OUT_F + lane] = fmaxf(acc0 * inv + b_gat[lane], 0.f);
    if (lane < 16)
        out[node * OUT_F + 32 + lane] = fmaxf(acc1 * inv + b_gat[32 + lane], 0.f);
}

// ---------------------------------------------------------------------------
// Stage 4: v2 = concat(residue_onehot[8], pssm[20], v1[4000])
// ---------------------------------------------------------------------------
__global__ void concat_kernel(const float* __restrict__ residue,
                              const float* __restrict__ pssm,
                              const float* __restrict__ v1,
                              float* __restrict__ v2) {
    const int i = blockIdx.x * blockDim.x + threadIdx.x;
    if (i >= 4028) return;
    float val;
    if      (i < 8)  val = residue[i];
    else if (i < 28) val = pssm[i - 8];
    else             val = v1[i - 28];
    v2[i] = val;
}

// ---------------------------------------------------------------------------
// Generic fp32 mat-vec: y = [relu](W @ x + b).  One wave per output row,
// vector cached in LDS, 128-bit weight loads, prefetch ahead of the stream.
// This kernel is the bandwidth roof of the whole net (W0: 183 MiB stream).
// cols must be a multiple of 4 (true for 12000/4028/800/400/200).
// ---------------------------------------------------------------------------
__global__ void matvec_kernel(const float* __restrict__ W,
                              const float* __restrict__ bias,
                              const float* __restrict__ x,
                              float* __restrict__ y,
                              int rows, int cols, int do_relu) {
    extern __shared__ float sx[];
    for (int i = threadIdx.x; i < cols; i += blockDim.x) sx[i] = x[i];
    __syncthreads();

    const int wave = blockIdx.x * (blockDim.x >> 5) + (threadIdx.x >> 5);
    const int lane = threadIdx.x & 31;
    if (wave >= rows) return;

    const float4* __restrict__ Wr = (const float4*)(W + (size_t)wave * cols);
    const float4* __restrict__ xr = (const float4*)sx;
    const int n4 = cols >> 2;

    float sum = 0.f;
    for (int i = lane; i < n4; i += 32) {
        if (i + 512 < n4) __builtin_prefetch(&Wr[i + 512], 0, 0);
        const float4 wv = Wr[i];
        const float4 xv = xr[i];
        sum += wv.x * xv.x + wv.y * xv.y + wv.z * xv.z + wv.w * xv.w;
    }
    for (int off = 16; off; off >>= 1) sum += __shfl_xor(sum, off, 32);

    if (lane == 0) {
        const float r = sum + bias[wave];
        y[wave] = do_relu ? fmaxf(r, 0.f) : r;
    }
}

// ---------------------------------------------------------------------------
extern "C" void kernel_launch(void* const* d_in, const int* in_sizes, int n_in,
                              void* d_out, int out_size, void* d_ws, size_t ws_size,
                              hipStream_t stream) {
    const float* x       = (const float*)d_in[0];
    const int*   ei      = (const int*)  d_in[1];
    /* d_in[2] edge_weight unused (matches reference) */
    const float* residue = (const float*)d_in[3];
    const float* pssm    = (const float*)d_in[4];
    const float* w_gat   = (const float*)d_in[5];
    const float* att_src = (const float*)d_in[6];
    const float* att_dst = (const float*)d_in[7];
    const float* b_gat   = (const float*)d_in[8];
    const float* W0 = (const float*)d_in[9];   const float* b0 = (const float*)d_in[10];
    const float* W1 = (const float*)d_in[11];  const float* b1 = (const float*)d_in[12];
    const float* W2 = (const float*)d_in[13];  const float* b2 = (const float*)d_in[14];
    const float* W3 = (const float*)d_in[15];  const float* b3 = (const float*)d_in[16];
    const float* W4 = (const float*)d_in[17];  const float* b4 = (const float*)d_in[18];
    float* out = (float*)d_out;

    float* ws  = (float*)d_ws;          // all offsets 16B-aligned
    float* h   = ws;                    // 12000  (pre-activation GAT features)
    float* a_s = ws + 12000;            // 250 (pad to 256)
    float* a_d = ws + 12256;            // 250 (pad to 256)
    float* v0  = ws + 12512;            // 12000  (relu(GAT) flattened)
    float* v1  = ws + 24512;            // 4000
    float* v2  = ws + 28512;            // 4028 (pad to 4032)
    float* v3  = ws + 32544;            // 800
    float* v4  = ws + 33344;            // 400
    float* v5  = ws + 33744;            // 200

    // Stage 1: WMMA fp32 GEMM (16 row-tiles x 3 col-tiles, one wave each)
    gat_gemm_wmma_kernel<<<48, 32, 0, stream>>>(x, w_gat, h);
    // Stage 2: attention scores
    att_score_kernel<<<1, 256, 0, stream>>>(h, att_src, att_dst, a_s, a_d);
    // Stage 3: deterministic segment softmax + aggregation (+bias, relu)
    gat_aggregate_kernel<<<32, 256, 0, stream>>>(ei, h, a_s, a_d, b_gat, v0);
    // Stage 4: W0 @ v0 — the 183 MiB bandwidth-bound stream (500 WGs in flight)
    matvec_kernel<<<500, 256, 12000 * 4, stream>>>(W0, b0, v0, v1, 4000, 12000, 1);
    // Stage 5: concat
    concat_kernel<<<16, 256, 0, stream>>>(residue, pssm, v1, v2);
    // Stages 6-9: remaining MLP
    matvec_kernel<<<100, 256, 4032 * 4, stream>>>(W1, b1, v2, v3, 800, 4028, 1);
    matvec_kernel<<< 50, 256,  800 * 4, stream>>>(W2, b2, v3, v4, 400,  800, 1);
    matvec_kernel<<< 25, 256,  400 * 4, stream>>>(W3, b3, v4, v5, 200,  400, 1);
    matvec_kernel<<<  1,  32,  200 * 4, stream>>>(W4, b4, v5, out,  1,  200, 0);
}